// GCN2LayerConcat2FCs_26560077758928
// MI455X (gfx1250) — compile-verified
//
#include <hip/hip_runtime.h>
#include <math.h>

#define N_NODES 64000
#define NODES_PER_GRAPH 1000
#define NUM_GRAPHS 64
#define HDIM 64

typedef __attribute__((ext_vector_type(2))) float v2f;
typedef __attribute__((ext_vector_type(8))) float v8f;

__device__ __forceinline__ v8f wmma_f32_4(v2f a, v2f b, v8f c) {
  // D = A(16x4) * B(4x16) + C(16x16), fp32, wave32
  return __builtin_amdgcn_wmma_f32_16x16x4_f32(
      /*neg_a=*/false, a, /*neg_b=*/false, b,
      /*c_mod=*/(short)0, c, /*reuse_a=*/false, /*reuse_b=*/false);
}

// ---------------------------------------------------------------- utilities
__global__ void fill_kernel(float* __restrict__ p, float v, int n) {
  int i = blockIdx.x * blockDim.x + threadIdx.x;
  if (i < n) p[i] = v;
}

__global__ void deg_kernel(const long long* __restrict__ col,
                           float* __restrict__ deg, int E) {
  int e = blockIdx.x * blockDim.x + threadIdx.x;
  if (e < E) atomicAdd(&deg[(int)col[e]], 1.0f);
}

__global__ void dinv_kernel(const float* __restrict__ deg,
                            float* __restrict__ dinv, int n) {
  int i = blockIdx.x * blockDim.x + threadIdx.x;
  if (i < n) {
    float d = deg[i];
    dinv[i] = d > 0.0f ? rsqrtf(d) : 0.0f;
  }
}

// ------------------------------------------------- node GEMM: H = X @ W
// X: [N_NODES x K] row-major, W: [K x 64] row-major, H: [N_NODES x 64].
// One wave per 16-row strip; 4 n-tiles of 16 held in 4 accumulators.
template <int K>
__global__ void node_gemm_kernel(const float* __restrict__ X,
                                 const float* __restrict__ W,
                                 float* __restrict__ Hout) {
  int wave = (blockIdx.x * blockDim.x + threadIdx.x) >> 5;
  int lane = threadIdx.x & 31;
  int m0 = wave << 4;                 // strip base row (grid sized exactly)
  int nlo = lane & 15;
  int koff = (lane >> 4) << 1;        // 0 for lanes 0-15, 2 for lanes 16-31
  const float* xr = X + (size_t)(m0 + nlo) * K;
  v8f acc0 = {}, acc1 = {}, acc2 = {}, acc3 = {};
#pragma unroll
  for (int k = 0; k < K; k += 4) {
    v2f a;
    a.x = xr[k + koff];
    a.y = xr[k + koff + 1];
    const float* w0 = W + (size_t)(k + koff) * HDIM + nlo;
    const float* w1 = w0 + HDIM;
    v2f b0, b1, b2, b3;
    b0.x = w0[0];  b0.y = w1[0];
    b1.x = w0[16]; b1.y = w1[16];
    b2.x = w0[32]; b2.y = w1[32];
    b3.x = w0[48]; b3.y = w1[48];
    acc0 = wmma_f32_4(a, b0, acc0);
    acc1 = wmma_f32_4(a, b1, acc1);
    acc2 = wmma_f32_4(a, b2, acc2);
    acc3 = wmma_f32_4(a, b3, acc3);
  }
  int mstore = m0 + ((lane >> 4) << 3);  // M = m0 + r + 8*(lane>=16)
#pragma unroll
  for (int r = 0; r < 8; ++r) {
    float* hr = Hout + (size_t)(mstore + r) * HDIM + nlo;
    hr[0]  = acc0[r];
    hr[16] = acc1[r];
    hr[32] = acc2[r];
    hr[48] = acc3[r];
  }
}

// -------------------------------------------- edge message scatter-add
// agg[col] += h[row] * dinv[row]*dinv[col]; one thread per (edge, 4 feats).
__global__ void scatter_kernel(const float* __restrict__ H,
                               const long long* __restrict__ row,
                               const long long* __restrict__ col,
                               const float* __restrict__ dinv,
                               float* __restrict__ agg, int E) {
  unsigned tid = blockIdx.x * blockDim.x + threadIdx.x;
  if (tid >= (unsigned)E * 16u) return;
  int e = (int)(tid >> 4);
  int f = (int)(tid & 15u) * 4;
  int r = (int)row[e];
  int c = (int)col[e];
  float nrm = dinv[r] * dinv[c];
  float4 hv = *(const float4*)(H + (size_t)r * HDIM + f);
  float* ap = agg + (size_t)c * HDIM + f;
  atomicAdd(ap + 0, hv.x * nrm);
  atomicAdd(ap + 1, hv.y * nrm);
  atomicAdd(ap + 2, hv.z * nrm);
  atomicAdd(ap + 3, hv.w * nrm);
}

// -------------------------------- self-loop + bias + tanh activation
__global__ void act_kernel(const float* __restrict__ agg,
                           const float* __restrict__ Hlin,
                           const float* __restrict__ dinv,
                           const float* __restrict__ bias,
                           float* __restrict__ out) {
  int idx = blockIdx.x * blockDim.x + threadIdx.x;
  if (idx < N_NODES * HDIM) {
    int node = idx >> 6;
    int f = idx & 63;
    float di = dinv[node];
    out[idx] = tanhf(agg[idx] + Hlin[idx] * di * di + bias[f]);
  }
}

// ------------------------------------------------ FC1: Z = Hc @ Wfc1
// Hc: [64 x 64000] (flat hact2), Wfc1: [64000 x 1000]. K split into chunks,
// partial sums accumulated with float atomics into Z [64 x 1000].
// Edge n-tile handled branchlessly: clamped (always-valid) address + 0/1
// scale, so the hot streaming loop has no exec-mask divergence.
#define FC1_KC 256
__global__ void fc1_kernel(const float* __restrict__ Hc,
                           const float* __restrict__ Wfc1,
                           float* __restrict__ Z) {
  int nt = blockIdx.x;                  // 0..62 (n-tile)
  int kc = blockIdx.y;                  // 0..249 (k-chunk)
  int lane = threadIdx.x & 31;
  int nlo = lane & 15;
  int n = nt * 16 + nlo;
  int koff = (lane >> 4) << 1;
  bool nok = n < NODES_PER_GRAPH;
  int nc = nok ? n : (NODES_PER_GRAPH - 1);   // clamped, always in-bounds
  float bscale = nok ? 1.0f : 0.0f;           // zero out-of-range columns
  const float* wp = Wfc1 + nc;
  const float* a0p = Hc + (size_t)nlo * 64000;
  v8f acc0 = {}, acc1 = {}, acc2 = {}, acc3 = {};
  int kbeg = kc * FC1_KC;
#pragma unroll 4
  for (int k = kbeg; k < kbeg + FC1_KC; k += 4) {
    int kk = k + koff;
    v2f b;
    b.x = wp[(size_t)kk * NODES_PER_GRAPH] * bscale;
    b.y = wp[(size_t)(kk + 1) * NODES_PER_GRAPH] * bscale;
    v2f a0, a1, a2, a3;
    a0.x = a0p[kk];                a0.y = a0p[kk + 1];
    a1.x = a0p[16 * 64000 + kk];   a1.y = a0p[16 * 64000 + kk + 1];
    a2.x = a0p[32 * 64000 + kk];   a2.y = a0p[32 * 64000 + kk + 1];
    a3.x = a0p[48 * 64000 + kk];   a3.y = a0p[48 * 64000 + kk + 1];
    acc0 = wmma_f32_4(a0, b, acc0);
    acc1 = wmma_f32_4(a1, b, acc1);
    acc2 = wmma_f32_4(a2, b, acc2);
    acc3 = wmma_f32_4(a3, b, acc3);
  }
  if (nok) {
    int mb = (lane >> 4) << 3;
#pragma unroll
    for (int r = 0; r < 8; ++r) {
      atomicAdd(&Z[(size_t)(mb + r) * NODES_PER_GRAPH + n], acc0[r]);
      atomicAdd(&Z[(size_t)(16 + mb + r) * NODES_PER_GRAPH + n], acc1[r]);
      atomicAdd(&Z[(size_t)(32 + mb + r) * NODES_PER_GRAPH + n], acc2[r]);
      atomicAdd(&Z[(size_t)(48 + mb + r) * NODES_PER_GRAPH + n], acc3[r]);
    }
  }
}

// ------------------------- FC2: out[g] = relu(Z[g]+bfc1) . Wfc2 + bfc2
__global__ void fc2_kernel(const float* __restrict__ Z,
                           const float* __restrict__ bfc1,
                           const float* __restrict__ Wfc2,
                           const float* __restrict__ bfc2,
                           float* __restrict__ out) {
  __shared__ float red[256];
  int g = blockIdx.x;
  int t = threadIdx.x;
  float s = 0.0f;
  for (int n = t; n < NODES_PER_GRAPH; n += 256)
    s += fmaxf(Z[(size_t)g * NODES_PER_GRAPH + n] + bfc1[n], 0.0f) * Wfc2[n];
  red[t] = s;
  __syncthreads();
  for (int w = 128; w > 0; w >>= 1) {
    if (t < w) red[t] += red[t + w];
    __syncthreads();
  }
  if (t == 0) out[g] = red[0] + bfc2[0];
}

// ---------------------------------------------------------------- driver
extern "C" void kernel_launch(void* const* d_in, const int* in_sizes, int n_in,
                              void* d_out, int out_size, void* d_ws, size_t ws_size,
                              hipStream_t stream) {
  const float* x        = (const float*)d_in[0];
  const long long* ei   = (const long long*)d_in[1];   // int64 [2, E]
  // d_in[2] = batch (unused: equal-size sorted blocks -> reshape)
  const float* W1   = (const float*)d_in[3];
  const float* b1   = (const float*)d_in[4];
  const float* W2   = (const float*)d_in[5];
  const float* b2   = (const float*)d_in[6];
  const float* Wfc1 = (const float*)d_in[7];
  const float* bfc1 = (const float*)d_in[8];
  const float* Wfc2 = (const float*)d_in[9];
  const float* bfc2 = (const float*)d_in[10];
  float* out = (float*)d_out;

  int E = in_sizes[1] / 2;
  const long long* row = ei;
  const long long* col = ei + E;

  float* ws   = (float*)d_ws;
  float* deg  = ws;                                  // 64000
  float* dinv = ws + 64000;                          // 64000
  float* B0   = ws + 128000;                         // linear h   [64000*64]
  float* B1   = B0 + (size_t)N_NODES * HDIM;         // aggregated [64000*64]
  float* B2   = B1 + (size_t)N_NODES * HDIM;         // activated  [64000*64]
  float* Z    = B2 + (size_t)N_NODES * HDIM;         // [64*1000]

  const int HN = N_NODES * HDIM;                     // 4,096,000
  const int ZN = NUM_GRAPHS * NODES_PER_GRAPH;       // 64,000

  // init: deg=1 (self-loops), agg=0, Z=0
  fill_kernel<<<(N_NODES + 255) / 256, 256, 0, stream>>>(deg, 1.0f, N_NODES);
  fill_kernel<<<(HN + 255) / 256, 256, 0, stream>>>(B1, 0.0f, HN);
  fill_kernel<<<(ZN + 255) / 256, 256, 0, stream>>>(Z, 0.0f, ZN);

  deg_kernel<<<(E + 255) / 256, 256, 0, stream>>>(col, deg, E);
  dinv_kernel<<<(N_NODES + 255) / 256, 256, 0, stream>>>(deg, dinv, N_NODES);

  unsigned sgrid = ((unsigned)E * 16u + 255u) / 256u;

  // ---- GCN layer 1 (K = 32)
  node_gemm_kernel<32><<<N_NODES / 16 / 8, 256, 0, stream>>>(x, W1, B0);
  scatter_kernel<<<sgrid, 256, 0, stream>>>(B0, row, col, dinv, B1, E);
  act_kernel<<<(HN + 255) / 256, 256, 0, stream>>>(B1, B0, dinv, b1, B2);

  // ---- GCN layer 2 (K = 64)
  node_gemm_kernel<64><<<N_NODES / 16 / 8, 256, 0, stream>>>(B2, W2, B0);
  fill_kernel<<<(HN + 255) / 256, 256, 0, stream>>>(B1, 0.0f, HN);
  scatter_kernel<<<sgrid, 256, 0, stream>>>(B0, row, col, dinv, B1, E);
  act_kernel<<<(HN + 255) / 256, 256, 0, stream>>>(B1, B0, dinv, b2, B2);

  // ---- FC head
  fc1_kernel<<<dim3(63, 64000 / FC1_KC), 32, 0, stream>>>(B2, Wfc1, Z);
  fc2_kernel<<<NUM_GRAPHS, 256, 0, stream>>>(Z, bfc1, Wfc2, bfc2, out);
}